// PowerIteration_43490838839883
// MI455X (gfx1250) — compile-verified
//
#include <hip/hip_runtime.h>
#include <math.h>

// Power iteration: v_{k+1} = normalize(M v_k), M 8192x8192 fp32, 50 iters.
// Memory-bound: 50 * 268MB = 13.4GB => ~575us floor at 23.3 TB/s HBM (L2
// holds 192MB of the 268MB matrix across passes). fp32 kept end-to-end
// (non-symmetric Gaussian M => trajectory precision-sensitive).
// Core: V_WMMA_F32_16X16X4_F32. Norm carried across iterations as a
// fixed-point u64 (deterministic integer atomics) so each iteration is a
// single kernel: w_{i+1} = (M w_i) * rsqrt(S_i), S_{i+1} += round(w²·2^32).

#define N 8192
#define MAX_ITER 50
#define KSLICE 1024              // K range per wave (8 waves x 1024 = 8192)
#define FIX_SCALE 4294967296.0   // 2^32

typedef float v2f __attribute__((ext_vector_type(2)));
typedef float v8f __attribute__((ext_vector_type(8)));

__global__ void pi_init(float* __restrict__ w, unsigned long long* __restrict__ acc) {
    int i = blockIdx.x * blockDim.x + threadIdx.x;
    if (i < N) w[i] = 1.0f;                       // v0 = ones
    if (blockIdx.x == 0 && threadIdx.x <= MAX_ITER) {
        // S_0 = sum(ones^2) = N, in 32.32 fixed point; later slots zeroed.
        acc[threadIdx.x] = (threadIdx.x == 0)
                               ? ((unsigned long long)N << 32)
                               : 0ull;
    }
}

// One block per 16-row tile (512 blocks), 8 waves per block, each wave owns a
// 1024-wide K slice. WMMA f32 16x16x4: A = 16x4 tile of M, B = w chunk
// replicated across all 16 columns; C accumulates row dot-products.
__global__ __launch_bounds__(256) void pi_step_wmma(
        const float* __restrict__ M,
        const float* __restrict__ win,
        float* __restrict__ wout,
        const unsigned long long* __restrict__ accIn,
        unsigned long long* __restrict__ accOut) {
    __shared__ float red[8][16];

    const int lane = threadIdx.x & 31;
    const int wave = threadIdx.x >> 5;     // 0..7
    const int half = lane >> 4;            // 0 or 1
    const int rowBase = blockIdx.x * 16;
    const int row = rowBase + (lane & 15);

    // 1/||w_prev|| from the previous iteration's fixed-point accumulator
    // (complete at kernel boundary; uniform scalar per wave).
    const float S = (float)((double)accIn[0] * (1.0 / FIX_SCALE));
    const float inv = 1.0f / sqrtf(S);

    // Pre-offset both streams by this wave's K base and this half-wave's
    // 2-element K phase, so the loop below has a compile-time trip count
    // (uniform scalar branch; EXEC stays all-1s through every WMMA).
    const float* __restrict__ Ap = M + (size_t)row * N + wave * KSLICE + 2 * half;
    const float* __restrict__ Bp = win + wave * KSLICE + 2 * half;

    v8f c = {0.f, 0.f, 0.f, 0.f, 0.f, 0.f, 0.f, 0.f};

    // A layout (16x4 f32): lanes 0-15 -> row=lane, K0/K1; lanes 16-31 -> row,
    // K2/K3 (lane pairs l,l+16 stream one row contiguously). B (4x16): same K
    // split per half-wave, replicated across N (broadcast load).
    for (int kk = 0; kk < KSLICE; kk += 32) {
        // One prefetch per 128B line (8 wmma steps), ~1KB ahead on A stream.
        __builtin_prefetch(Ap + kk + 256, 0, 1);
        #pragma unroll
        for (int j = 0; j < 8; ++j) {
            const int ka = kk + 4 * j;
            v2f a = *(const v2f*)(Ap + ka);
            v2f b = *(const v2f*)(Bp + ka);
            c = __builtin_amdgcn_wmma_f32_16x16x4_f32(
                    /*neg_a=*/false, a, /*neg_b=*/false, b,
                    /*c_mod=*/(short)0, c, /*reuse_a=*/false, /*reuse_b=*/false);
        }
    }

    // D columns identical; lane 0 holds rows 0..7 (VGPR 0..7), lane 16 holds
    // rows 8..15. Deposit per-wave partials, reduce deterministically.
    if (lane == 0) {
        #pragma unroll
        for (int i = 0; i < 8; ++i) red[wave][i] = c[i];
    }
    if (lane == 16) {
        #pragma unroll
        for (int i = 0; i < 8; ++i) red[wave][8 + i] = c[i];
    }
    __syncthreads();

    if (threadIdx.x < 16) {
        float s = 0.0f;
        #pragma unroll
        for (int w = 0; w < 8; ++w) s += red[w][threadIdx.x];
        const float val = s * inv;             // w_{i+1} = (M w_i)/||w_i||
        wout[rowBase + threadIdx.x] = val;
        // Deterministic norm accumulation: integer atomics are associative.
        const unsigned long long q =
            (unsigned long long)((double)(val * val) * FIX_SCALE);
        atomicAdd(accOut, q);
    }
}

__global__ void pi_finalize(const float* __restrict__ w,
                            const unsigned long long* __restrict__ acc,
                            float* __restrict__ out) {
    int i = blockIdx.x * blockDim.x + threadIdx.x;
    const float S = (float)((double)acc[0] * (1.0 / FIX_SCALE));
    const float inv = 1.0f / sqrtf(S);
    if (i < N) out[i] = w[i] * inv;
}

extern "C" void kernel_launch(void* const* d_in, const int* in_sizes, int n_in,
                              void* d_out, int out_size, void* d_ws, size_t ws_size,
                              hipStream_t stream) {
    (void)in_sizes; (void)n_in; (void)out_size; (void)ws_size;

    const float* M = (const float*)d_in[0];
    float* out = (float*)d_out;

    float* wA = (float*)d_ws;                               // N floats
    float* wB = wA + N;                                     // N floats
    unsigned long long* acc = (unsigned long long*)(wB + N); // MAX_ITER+1 u64

    pi_init<<<N / 256, 256, 0, stream>>>(wA, acc);

    for (int it = 0; it < MAX_ITER; ++it) {
        const float* win = (it & 1) ? wB : wA;
        float* wout      = (it & 1) ? wA : wB;
        pi_step_wmma<<<N / 16, 256, 0, stream>>>(M, win, wout,
                                                 acc + it, acc + it + 1);
    }

    // w_50 lives in wA (50 even); scale by final norm into d_out.
    pi_finalize<<<N / 256, 256, 0, stream>>>((MAX_ITER & 1) ? wB : wA,
                                             acc + MAX_ITER, out);
}